// SpanRepresentation_47742856462632
// MI455X (gfx1250) — compile-verified
//
#include <hip/hip_runtime.h>
#include <stdint.h>

typedef float v4f __attribute__((ext_vector_type(4)));

#define B_    16
#define L_    512
#define D_    768
#define W_    8
#define WD_   64
#define NS_   4068      // sum_{w=1..8} (512-w+1)
#define OUTD_ 1600      // 768 + 768 + 64
#define ROWS_ 16        // output rows (spans) per block
#define F4_PER_ROW 400  // 1600 / 4
#define F4_X  192       // 768 / 4

__global__ __launch_bounds__(256)
void span_rep_kernel(const float* __restrict__ x,
                     const float* __restrict__ we,
                     float* __restrict__ out) {
    // width_emb staged in LDS: 8 rows x 64 floats = 2 KB
    __shared__ v4f lds_we[W_ * WD_ / 4];   // 128 x 16B
    const int t = threadIdx.x;

    // ---- CDNA5 async global->LDS DMA: stage width_emb (2KB), 8B per lane ----
    {
        unsigned lds_off = (unsigned)(size_t)(&lds_we[0]) + (unsigned)t * 8u;
        unsigned g_off   = (unsigned)t * 8u;
        asm volatile("global_load_async_to_lds_b64 %0, %1, %2"
                     :
                     : "v"(lds_off), "v"(g_off), "s"(we)
                     : "memory");
        asm volatile("s_wait_asynccnt 0" ::: "memory");
    }
    __syncthreads();

    const int base_row = blockIdx.x * ROWS_;

    #pragma unroll 1
    for (int rl = 0; rl < ROWS_; ++rl) {
        const int row = base_row + rl;        // < 65088, uniform per block
        const int b   = row / NS_;
        const int s   = row - b * NS_;

        // width bucket (uniform -> scalar branches)
        int wi, so;
        if      (s < 512)  { wi = 0; so = 0;    }
        else if (s < 1023) { wi = 1; so = 512;  }
        else if (s < 1533) { wi = 2; so = 1023; }
        else if (s < 2042) { wi = 3; so = 1533; }
        else if (s < 2550) { wi = 4; so = 2042; }
        else if (s < 3057) { wi = 5; so = 2550; }
        else if (s < 3563) { wi = 6; so = 3057; }
        else               { wi = 7; so = 3563; }
        const int st = s - so;
        const int en = st + wi;

        const v4f* __restrict__ xs = (const v4f*)(x + ((size_t)b * L_ + st) * D_);
        const v4f* __restrict__ xe = (const v4f*)(x + ((size_t)b * L_ + en) * D_);
        const v4f* __restrict__ wv = &lds_we[wi * (WD_ / 4)];
        v4f* __restrict__ orow     = (v4f*)(out + (size_t)row * OUTD_);

        // iteration 1: c = t in [0, 256): h_start (c<192) or h_end.
        // Both sides are GLOBAL -> compiler's select-address merge gives one
        // global_load_b128, which is ideal.
        {
            const int c = t;
            v4f v;
            if (c < F4_X) v = xs[c];
            else          v = xe[c - F4_X];
            __builtin_nontemporal_store(v, &orow[c]);
        }
        // iteration 2: c = t + 256 in [256, 400): h_end (c<384) or width (LDS).
        // Load BOTH with always-in-bounds indices and select the VALUE, so the
        // compiler cannot merge a global pointer with an LDS pointer into a
        // generic/FLAT load. Keeps global_load_b128 + ds_load_b128 paths.
        {
            const int c = t + 256;
            if (c < F4_PER_ROW) {
                const int gi = (c < 2 * F4_X ? c : (2 * F4_X - 1)) - F4_X; // clamp to row
                const int li = (c - 2 * F4_X) & 15;                        // mask into row
                v4f vg = xe[gi];            // global_load_b128
                v4f vw = wv[li];            // ds_load_b128
                v4f v  = (c < 2 * F4_X) ? vg : vw;   // v_cndmask x4
                __builtin_nontemporal_store(v, &orow[c]);
            }
        }
    }
}

extern "C" void kernel_launch(void* const* d_in, const int* in_sizes, int n_in,
                              void* d_out, int out_size, void* d_ws, size_t ws_size,
                              hipStream_t stream) {
    const float* x  = (const float*)d_in[0];   // (16, 512, 768) f32
    const float* we = (const float*)d_in[1];   // (8, 64) f32
    // d_in[2] = batch_max_seq_len scalar (512), fixed by setup
    float* out = (float*)d_out;                // (16, 4068, 1600) f32

    const int rows   = B_ * NS_;     // 65088
    const int blocks = rows / ROWS_; // 4068 (exact)
    span_rep_kernel<<<blocks, 256, 0, stream>>>(x, we, out);
}